// LCMReconstruction_45131516346969
// MI455X (gfx1250) — compile-verified
//
#include <hip/hip_runtime.h>
#include <hip/hip_bf16.h>
#include <math.h>

// ---------------------------------------------------------------------------
// LCM spectral reconstruction, MI455X (gfx1250, wave32).
// DFT over the needed 1024 fftshifted bins is expressed as a complex GEMM
// (256x4096)x(4096x1024) executed with V_WMMA_F32_16X16X4_F32 (exact fp32).
// Working set (~52MB of workspace) fits entirely in the 192MB L2.
// ---------------------------------------------------------------------------

#define BATCH     256
#define TLEN      4096
#define NP        1024          // START_IDX(2099) - END_IDX(1075)
#define ENDIDX    1075
#define NB_OFF    26
#define NB_DIFF   27
#define BT        (BATCH*TLEN)  // 1048576
#define BNP       (BATCH*NP)    // 262144
#define PSTRIDE   128

typedef __attribute__((ext_vector_type(2))) float v2f;
typedef __attribute__((ext_vector_type(8))) float v8f;

static constexpr float kPI   = 3.14159265358979323846f;
static constexpr float kDT   = 0.0005f;
static constexpr float kFRM  = 42.577f * 3.0f * 0.03f;   // FR_METAB
static constexpr float kFRMM = 42.577f * 3.0f * 0.05f;   // FR_MM
static constexpr float kSqrt5000 = 70.71067811865476f;

// --------------------------- parameter denorm ------------------------------
// P layout per batch (stride 128 floats):
//  [0..26]  amp (DIFF; OFF uses first 26)   [27] ph0*360  [28] ph1*10
//  [29]     lbG*sqrt(5000)
//  [32..58] lbL (n<18: *10, else *50)
//  [64..90] fr  (n<18: *FR_METAB, else *FR_MM)
//  [96..101] bl_OFF   [104..109] bl_DIFF
__global__ void prep_kernel(const float* __restrict__ pred,
                            const float* __restrict__ lv,
                            const float* __restrict__ hv,
                            float* __restrict__ P) {
  int b = blockIdx.x;
  int j = threadIdx.x;
  if (j >= 96) return;
  float x = pred[b * 96 + j];
  float d = x * (hv[j] - lv[j]) + lv[j];
  float* pb = P + b * PSTRIDE;
  if (j < 27)         pb[j] = d;
  else if (j == 27)   pb[27] = d * 360.0f;
  else if (j == 28)   pb[28] = d * 10.0f;
  else if (j == 29)   pb[29] = d * kSqrt5000;
  else if (j < 57)  { int n = j - 30; pb[32 + n] = d * (n < 18 ? 10.0f : 50.0f); }
  else if (j < 84)  { int n = j - 57; pb[64 + n] = d * (n < 18 ? kFRM : kFRMM); }
  else              { int c = j - 84; pb[((c & 1) ? 104 : 96) + (c >> 1)] = d; }
}

// --------------------------- DFT twiddle build -----------------------------
// W[t][p] = exp(-2*pi*i * k_p * t / 4096),  k_p = (ENDIDX + p + 2048) & 4095
// (fftshift folded into k_p). Exact integer angle reduction mod 4096.
__global__ void wbuild_kernel(float* __restrict__ Wre, float* __restrict__ Wim) {
  int idx = blockIdx.x * 256 + threadIdx.x;     // idx = t*NP + p
  int t = idx >> 10;
  int p = idx & (NP - 1);
  int k = (ENDIDX + p + TLEN / 2) & (TLEN - 1);
  int m = (k * t) & (TLEN - 1);                 // k*t < 2^25, exact in i32
  float ang = -2.0f * kPI * (float)m * (1.0f / (float)TLEN);
  float s, c;
  __sincosf(ang, &s, &c);
  Wre[idx] = c;
  Wim[idx] = s;
}

// --------------------------- FID synthesis ---------------------------------
// fid planes layout: [0]=OFF_re [1]=OFF_im [2]=DIFF_re [3]=DIFF_im, each B*T.
__global__ void fid_kernel(const float* __restrict__ P,
                           const float* __restrict__ bOr, const float* __restrict__ bOi,
                           const float* __restrict__ bDr, const float* __restrict__ bDi,
                           float* __restrict__ fid) {
  __shared__ float sp[PSTRIDE];
  int b = blockIdx.x >> 4;                        // T/256 = 16 chunks
  int t = ((blockIdx.x & 15) << 8) + threadIdx.x;
  if (threadIdx.x < PSTRIDE) sp[threadIdx.x] = P[b * PSTRIDE + threadIdx.x];
  __syncthreads();

  float tt = (float)t * kDT;
  float aOr = 0.f, aOi = 0.f, aDr = 0.f, aDi = 0.f;
  for (int n = 0; n < NB_DIFF; ++n) {
    float c = sp[n] * __expf(-kPI * sp[32 + n] * tt);   // amp * Lorentz decay
    float s, co;
    __sincosf(2.0f * kPI * sp[64 + n] * tt, &s, &co);   // freq phase
    float wr = c * co, wi = c * s;
    float br = bDr[t * NB_DIFF + n], bi = bDi[t * NB_DIFF + n];
    aDr = fmaf(br, wr, fmaf(-bi, wi, aDr));
    aDi = fmaf(br, wi, fmaf( bi, wr, aDi));
    if (n < NB_OFF) {
      float br2 = bOr[t * NB_OFF + n], bi2 = bOi[t * NB_OFF + n];
      aOr = fmaf(br2, wr, fmaf(-bi2, wi, aOr));
      aOi = fmaf(br2, wi, fmaf( bi2, wr, aOi));
    }
  }
  float g    = sp[29] * kPI * tt;                       // Gaussian window
  float winG = __expf(-g * g);
  float s0, c0;
  __sincosf(2.0f * kPI * sp[27] * (1.0f / 360.0f), &s0, &c0);  // ph0
  float mr = winG * c0, mi = winG * s0;
  int idx = b * TLEN + t;
  fid[idx]          = aOr * mr - aOi * mi;
  fid[BT + idx]     = aOr * mi + aOi * mr;
  fid[2 * BT + idx] = aDr * mr - aDi * mi;
  fid[3 * BT + idx] = aDr * mi + aDi * mr;
}

// --------------------------- WMMA DFT-GEMM ---------------------------------
// One wave per 16x16 output tile; 2 branches * 16 M-tiles * 64 N-tiles = 2048
// waves. Complex GEMM: Dre = Ar*Wre - Ai*Wim, Dim = Ar*Wim + Ai*Wre.
// A-fragment (fp32 16x4): lane m=L%16, VGPR0 = K=(L<16?0:2), VGPR1 = K+1.
// Imag-A negation done in VALU (fp WMMA NEG[1:0] must be zero).
__global__ void dft_gemm_kernel(const float* __restrict__ fid,
                                const float* __restrict__ Wre,
                                const float* __restrict__ Wim,
                                float* __restrict__ spec) {
  int gtid   = blockIdx.x * blockDim.x + threadIdx.x;
  int wave   = gtid >> 5;
  int lane   = gtid & 31;
  int branch = wave >> 10;
  int tid    = wave & 1023;
  int mt     = tid >> 6;     // 0..15  (rows of 16)
  int nt     = tid & 63;     // 0..63  (cols of 16)

  const float* Ar = fid + branch * 2 * BT;
  const float* Ai = Ar + BT;
  int mrow  = mt * 16 + (lane & 15);
  int ncol  = nt * 16 + (lane & 15);
  int khalf = (lane >> 4) << 1;                 // 0 or 2
  const float* arp = Ar + mrow * TLEN + khalf;
  const float* aip = Ai + mrow * TLEN + khalf;
  const float* wrp = Wre + khalf * NP + ncol;
  const float* wip = Wim + khalf * NP + ncol;

  v8f cre = {0.f, 0.f, 0.f, 0.f, 0.f, 0.f, 0.f, 0.f};
  v8f cim = {0.f, 0.f, 0.f, 0.f, 0.f, 0.f, 0.f, 0.f};
  for (int k = 0; k < TLEN; k += 4) {
    v2f ar = *(const v2f*)(arp + k);            // A[m][k+khalf .. +1]
    v2f ai = *(const v2f*)(aip + k);
    v2f wr, wi;                                 // B[k+khalf][n], B[k+khalf+1][n]
    wr.x = wrp[k * NP];  wr.y = wrp[(k + 1) * NP];
    wi.x = wip[k * NP];  wi.y = wip[(k + 1) * NP];
    v2f ain = -ai;
    cre = __builtin_amdgcn_wmma_f32_16x16x4_f32(false, ar,  false, wr, (short)0, cre, false, false);
    cre = __builtin_amdgcn_wmma_f32_16x16x4_f32(false, ain, false, wi, (short)0, cre, false, false);
    cim = __builtin_amdgcn_wmma_f32_16x16x4_f32(false, ar,  false, wi, (short)0, cim, false, false);
    cim = __builtin_amdgcn_wmma_f32_16x16x4_f32(false, ai,  false, wr, (short)0, cim, false, false);
  }
  // D layout: VGPR r -> (M = r + (lane>=16 ? 8 : 0), N = lane&15)
  float* sre = spec + branch * 2 * BNP;
  float* sim = sre + BNP;
  int rbase = mt * 16 + ((lane >> 4) << 3);
  int col   = nt * 16 + (lane & 15);
#pragma unroll
  for (int r = 0; r < 8; ++r) {
    sre[(rbase + r) * NP + col] = cre[r];
    sim[(rbase + r) * NP + col] = cim[r];
  }
}

// --------------------- ph1 rotation + baseline + normalize -----------------
__global__ void post_kernel(const float* __restrict__ spec,
                            const float* __restrict__ P,
                            const float* __restrict__ ppm,
                            float* __restrict__ out) {
  int b      = blockIdx.x >> 1;
  int branch = blockIdx.x & 1;
  int p      = threadIdx.x;
  const float* sre = spec + branch * 2 * BNP + b * NP;
  const float* sim = sre + BNP;

  float ph1   = P[b * PSTRIDE + 28];
  float theta = 2.0f * kPI * (ph1 * (1.0f / 360.0f)) * ppm[ENDIDX + p];
  float s, c;
  __sincosf(theta, &s, &c);
  float val = sre[p] * c - sim[p] * s;          // Re(spec * p1)

  float x = -1.0f + 2.0f * (float)p / (float)(NP - 1);
  const float* bl = P + b * PSTRIDE + (branch ? 104 : 96);
  float poly = bl[0];
  poly = poly * x + bl[1];
  poly = poly * x + bl[2];
  poly = poly * x + bl[3];
  poly = poly * x + bl[4];
  poly = poly * x + bl[5];
  val += poly;

  __shared__ float red[32];
  float m = fabsf(val);
  for (int off = 16; off > 0; off >>= 1)
    m = fmaxf(m, __shfl_xor(m, off, 32));
  if ((p & 31) == 0) red[p >> 5] = m;
  __syncthreads();
  if (p < 32) {
    float mm = red[p];
    for (int off = 16; off > 0; off >>= 1)
      mm = fmaxf(mm, __shfl_xor(mm, off, 32));
    if (p == 0) red[0] = mm;
  }
  __syncthreads();
  float inv = 1.0f / red[0];
  out[((size_t)b * NP + p) * 2 + branch] = val * inv;   // stack OFF/DIFF last
}

// ---------------------------------------------------------------------------
extern "C" void kernel_launch(void* const* d_in, const int* in_sizes, int n_in,
                              void* d_out, int out_size, void* d_ws, size_t ws_size,
                              hipStream_t stream) {
  (void)in_sizes; (void)n_in; (void)out_size; (void)ws_size;
  const float* pred = (const float*)d_in[0];
  const float* bOr  = (const float*)d_in[1];
  const float* bOi  = (const float*)d_in[2];
  const float* bDr  = (const float*)d_in[3];
  const float* bDi  = (const float*)d_in[4];
  const float* ppm  = (const float*)d_in[5];
  const float* lv   = (const float*)d_in[6];
  const float* hv   = (const float*)d_in[7];

  // Workspace map (floats): total ~13.7M floats (~55 MB) -> fits 192MB L2.
  float* ws   = (float*)d_ws;
  float* P    = ws;                                  // BATCH*128      = 32768
  float* fid  = P   + (size_t)BATCH * PSTRIDE;       // 4*BT           = 4194304
  float* Wre  = fid + (size_t)4 * BT;                // TLEN*NP        = 4194304
  float* Wim  = Wre + (size_t)TLEN * NP;             // TLEN*NP        = 4194304
  float* spec = Wim + (size_t)TLEN * NP;             // 4*BNP          = 1048576
  float* out  = (float*)d_out;

  prep_kernel<<<BATCH, 96, 0, stream>>>(pred, lv, hv, P);
  wbuild_kernel<<<(TLEN * NP) / 256, 256, 0, stream>>>(Wre, Wim);
  fid_kernel<<<BATCH * (TLEN / 256), 256, 0, stream>>>(P, bOr, bOi, bDr, bDi, fid);
  dft_gemm_kernel<<<256, 256, 0, stream>>>(fid, Wre, Wim, spec);  // 2048 waves
  post_kernel<<<BATCH * 2, NP, 0, stream>>>(spec, P, ppm, out);
}